// Model_36893769073247
// MI455X (gfx1250) — compile-verified
//
#include <hip/hip_runtime.h>

typedef __attribute__((ext_vector_type(2))) float v2f;
typedef __attribute__((ext_vector_type(8))) float v8f;

// ---------------------------------------------------------------------------
// lower_bound over sorted int array (dst arrays are sorted in the reference)
// ---------------------------------------------------------------------------
__device__ __forceinline__ int lower_bound_i(const int* __restrict__ a, int n, int key) {
    int lo = 0, hi = n;
    while (lo < hi) {
        int mid = (lo + hi) >> 1;
        if (a[mid] < key) lo = mid + 1; else hi = mid;
    }
    return lo;
}

// ---------------------------------------------------------------------------
// Segment-mean aggregation: out[d, :] = mean over edges e with dst[e]==d of
// feat[src[e], :].  One wave32 per destination node.  F floats per row,
// F/128 float4 per lane.  Edge indices are loaded cooperatively (one per
// lane) and broadcast with __shfl so each src[e] is fetched once per wave.
// This is the HBM hot spot (~1.5 GB total -> ~65 us floor at 23.3 TB/s).
// ---------------------------------------------------------------------------
template <int F>
__global__ void seg_mean_kernel(const float* __restrict__ feat,
                                const int*   __restrict__ src,
                                const int*   __restrict__ dst,
                                float*       __restrict__ out,
                                int n_edges, int n_dst) {
    const int lane = threadIdx.x & 31;
    const int wave = (int)((blockIdx.x * blockDim.x + threadIdx.x) >> 5);
    if (wave >= n_dst) return;

    const int lo = lower_bound_i(dst, n_edges, wave);
    const int hi = lower_bound_i(dst, n_edges, wave + 1);

    constexpr int NV = F / 128;            // float4 chunks per lane
    float4 acc[NV];
#pragma unroll
    for (int i = 0; i < NV; ++i) acc[i] = make_float4(0.f, 0.f, 0.f, 0.f);

    for (int base = lo; base < hi; base += 32) {
        const int e     = base + lane;
        const int mySrc = (e < hi) ? src[e] : 0;
        const int cnt   = min(32, hi - base);
        for (int j = 0; j < cnt; ++j) {
            const int s = __shfl(mySrc, j, 32);          // wave-uniform broadcast
            const float4* row = (const float4*)(feat + (size_t)s * F);
#pragma unroll
            for (int i = 0; i < NV; ++i) {
                float4 v = row[lane + 32 * i];           // global_load_b128, coalesced
                acc[i].x += v.x; acc[i].y += v.y; acc[i].z += v.z; acc[i].w += v.w;
            }
        }
    }

    const float inv = 1.0f / (float)max(hi - lo, 1);
    float4* orow = (float4*)(out + (size_t)wave * F);
#pragma unroll
    for (int i = 0; i < NV; ++i) {
        acc[i].x *= inv; acc[i].y *= inv; acc[i].z *= inv; acc[i].w *= inv;
        orow[lane + 32 * i] = acc[i];
    }
}

// ---------------------------------------------------------------------------
// One accumulation pass of A(16xK) @ B(Kx16) using V_WMMA_F32_16X16X4_F32.
// N, K are compile-time so every load offset is a constant that folds into
// the 24-bit IOFFSET of the global_load (no per-iteration address math).
// Ap already points at A[row, khi]; Bp at B[khi, colC].
// ---------------------------------------------------------------------------
template <int N, int K>
__device__ __forceinline__ v8f wmma_pass(const float* __restrict__ Ap,
                                         const float* __restrict__ Bp,
                                         v8f acc) {
#pragma unroll
    for (int k = 0; k < K; k += 4) {
        v2f a, b;
        a.x = Ap[k];                 // contiguous pair -> global_load_b64 + imm off
        a.y = Ap[k + 1];
        b.x = Bp[(size_t)k * N];     // imm offsets: k*N*4 bytes (fits 24-bit)
        b.y = Bp[(size_t)(k + 1) * N];
        acc = __builtin_amdgcn_wmma_f32_16x16x4_f32(
            false, a, false, b, (short)0, acc, false, false);
    }
    return acc;
}

// ---------------------------------------------------------------------------
// Fused SAGE GEMM:  C[M,N] = act( A_self @ B_self + A_neigh @ B_neigh + bias )
// ldA = K, ldB = ldC = N (row-major).  One wave32 = one 16x16 C tile.
// Per ISA 16x4 f32 fragment layout: lanes 0-15 hold (M=lane, K=k,k+1),
// lanes 16-31 hold (M=lane-16, K=k+2,k+3); C VGPR j -> row j (+8 high lanes).
// For N=47 the out-of-range column is *clamped* for loads (stays in-bounds,
// no divergence, EXEC stays all-1s through every WMMA) and only the final
// store is guarded — garbage in an unstored D column is harmless because
// column n of D depends only on column n of B.
// ---------------------------------------------------------------------------
template <bool RELU, bool NGUARD, int N, int K>
__global__ void sage_gemm_wmma(const float* __restrict__ A_self,
                               const float* __restrict__ A_neigh,
                               const float* __restrict__ B_self,
                               const float* __restrict__ B_neigh,
                               const float* __restrict__ bias,
                               float*       __restrict__ C,
                               int M) {
    constexpr int nTilesN = (N + 15) / 16;
    const int lane = threadIdx.x & 31;
    const int tile = (int)(blockIdx.x * (blockDim.x >> 5) + (threadIdx.x >> 5));
    const int nTilesM = M >> 4;                 // M is a multiple of 16 here
    if (tile >= nTilesM * nTilesN) return;

    const int tm = tile / nTilesN;
    const int tn = tile - tm * nTilesN;

    const int col  = tn * 16 + (lane & 15);
    const int colC = NGUARD ? (col < N ? col : N - 1) : col;   // clamp, not mask
    const int arow = tm * 16 + (lane & 15);
    const int khi  = (lane >> 4) << 1;          // 0 for lanes 0-15, 2 for 16-31

    v8f acc = {0.f, 0.f, 0.f, 0.f, 0.f, 0.f, 0.f, 0.f};
    acc = wmma_pass<N, K>(A_self  + (size_t)arow * K + khi,
                          B_self  + (size_t)khi * N + colC, acc);
    acc = wmma_pass<N, K>(A_neigh + (size_t)arow * K + khi,
                          B_neigh + (size_t)khi * N + colC, acc);

    const float bv    = bias[colC];
    const int   rbase = tm * 16 + ((lane >> 4) << 3);
#pragma unroll
    for (int j = 0; j < 8; ++j) {
        float v = acc[j] + bv;
        if (RELU) v = fmaxf(v, 0.f);
        if (!NGUARD || col < N) C[(size_t)(rbase + j) * N + col] = v;
    }
}

// ---------------------------------------------------------------------------
// Problem constants (from the reference's setup_inputs)
// ---------------------------------------------------------------------------
static constexpr int kNDst1 = 50000;
static constexpr int kNDst2 = 10000;
static constexpr int kInF   = 128;
static constexpr int kHF    = 256;
static constexpr int kNCls  = 47;

extern "C" void kernel_launch(void* const* d_in, const int* in_sizes, int n_in,
                              void* d_out, int out_size, void* d_ws, size_t ws_size,
                              hipStream_t stream) {
    const float* x        = (const float*)d_in[0];
    const float* W_self1  = (const float*)d_in[1];
    const float* W_neigh1 = (const float*)d_in[2];
    const float* b1       = (const float*)d_in[3];
    const float* W_self2  = (const float*)d_in[4];
    const float* W_neigh2 = (const float*)d_in[5];
    const float* b2       = (const float*)d_in[6];
    const int*   src1     = (const int*)d_in[7];
    const int*   dst1     = (const int*)d_in[8];
    const int*   src2     = (const int*)d_in[9];
    const int*   dst2     = (const int*)d_in[10];
    const int    E1       = in_sizes[7];
    const int    E2       = in_sizes[9];

    // Workspace layout (all fully overwritten every call -> deterministic)
    float* h_neigh1 = (float*)d_ws;                               // 50000*128 f32
    float* h1       = h_neigh1 + (size_t)kNDst1 * kInF;           // 50000*256 f32
    float* h_neigh2 = h1       + (size_t)kNDst1 * kHF;            // 10000*256 f32
    (void)ws_size; (void)n_in; (void)out_size;

    // ---- Layer 1: mean aggregation of 128-dim src features over 2M edges ----
    {
        const int threads = kNDst1 * 32;
        seg_mean_kernel<kInF><<<(threads + 255) / 256, 256, 0, stream>>>(
            x, src1, dst1, h_neigh1, E1, kNDst1);
    }
    // ---- Layer 1: fused dual GEMM + bias + relu (FP32 WMMA) ----
    {
        const int tiles = (kNDst1 / 16) * (kHF / 16);   // 3125 * 16 = 50000
        sage_gemm_wmma<true, false, kHF, kInF><<<(tiles + 7) / 8, 256, 0, stream>>>(
            x, h_neigh1, W_self1, W_neigh1, b1, h1, kNDst1);
    }
    // ---- Layer 2: mean aggregation of 256-dim hidden features over 400K edges ----
    {
        const int threads = kNDst2 * 32;
        seg_mean_kernel<kHF><<<(threads + 255) / 256, 256, 0, stream>>>(
            h1, src2, dst2, h_neigh2, E2, kNDst2);
    }
    // ---- Layer 2: fused dual GEMM + bias (N = 47, clamped loads + guarded store) ----
    {
        const int tiles = (kNDst2 / 16) * ((kNCls + 15) / 16);   // 625 * 3 = 1875
        sage_gemm_wmma<false, true, kNCls, kHF><<<(tiles + 7) / 8, 256, 0, stream>>>(
            h1, h_neigh2, W_self2, W_neigh2, b2, (float*)d_out, kNDst2);
    }
}